// SwitchGating_33921651704036
// MI455X (gfx1250) — compile-verified
//
#include <hip/hip_runtime.h>

typedef __attribute__((ext_vector_type(2))) float v2f;
typedef __attribute__((ext_vector_type(4))) float v4f;
typedef __attribute__((ext_vector_type(8))) float v8f;

#define G_  4
#define S_  2048
#define M_  1024
#define E_  64
#define CAP_ 160
#define GS_ (G_ * S_)
static const size_t GSEC_ = (size_t)GS_ * E_ * CAP_;   // 83,886,080

// ---------------------------------------------------------------------------
// Kernel 1: logits GEMM via V_WMMA_F32_16X16X4_F32 + softmax + argmax + gate
//           + per-(g,e) softmax-prob sums (for aux loss density proxy).
// Block = 128 threads = 4 waves; each wave computes a 16-token x 64-expert
// logits tile (4 accumulators, K stepped by 4). Block covers 64 tokens.
// ---------------------------------------------------------------------------
__global__ __launch_bounds__(128) void k_gate(const float* __restrict__ inp,
                                              const float* __restrict__ W,
                                              int*   __restrict__ expert_out,
                                              float* __restrict__ gate_out,
                                              float* __restrict__ proxy_sum) {
  __shared__ float lds_logits[64 * 64];   // 64 tokens x 64 experts (16 KB)
  __shared__ float lds_max[64];
  __shared__ float lds_rsum[64];

  const int tid  = threadIdx.x;
  const int wave = tid >> 5;
  const int lane = tid & 31;
  const int blockTok = blockIdx.x * 64;
  const int tileTok  = blockTok + wave * 16;

  // A-matrix layout (f32 16x4): lane holds row M = lane%16; VGPR0/1 hold
  // K = koff, koff+1 with koff = (lane>=16 ? 2 : 0).
  const int mrow = lane & 15;
  const int koff = (lane >> 4) * 2;
  const int ncol = lane & 15;             // B/C column within 16-wide tile
  const float* Arow = inp + (size_t)(tileTok + mrow) * M_;

  v8f acc0 = {}, acc1 = {}, acc2 = {}, acc3 = {};
  for (int k0 = 0; k0 < M_; k0 += 4) {
    v2f a;
    a.x = Arow[k0 + koff];
    a.y = Arow[k0 + koff + 1];
    const float* Wk0 = W + (size_t)(k0 + koff) * E_;   // W[k][e], row-major
    const float* Wk1 = Wk0 + E_;
    v2f b0; b0.x = Wk0[ 0 + ncol]; b0.y = Wk1[ 0 + ncol];
    v2f b1; b1.x = Wk0[16 + ncol]; b1.y = Wk1[16 + ncol];
    v2f b2; b2.x = Wk0[32 + ncol]; b2.y = Wk1[32 + ncol];
    v2f b3; b3.x = Wk0[48 + ncol]; b3.y = Wk1[48 + ncol];
    acc0 = __builtin_amdgcn_wmma_f32_16x16x4_f32(false, a, false, b0, (short)0, acc0, false, false);
    acc1 = __builtin_amdgcn_wmma_f32_16x16x4_f32(false, a, false, b1, (short)0, acc1, false, false);
    acc2 = __builtin_amdgcn_wmma_f32_16x16x4_f32(false, a, false, b2, (short)0, acc2, false, false);
    acc3 = __builtin_amdgcn_wmma_f32_16x16x4_f32(false, a, false, b3, (short)0, acc3, false, false);
  }

  // C/D layout: VGPR r, lanes 0-15 -> M=r, lanes 16-31 -> M=8+r; N = lane%16.
  const int rbase = (lane < 16) ? 0 : 8;
  #pragma unroll
  for (int r = 0; r < 8; ++r) {
    const int row = wave * 16 + r + rbase;
    lds_logits[row * 64 +  0 + ncol] = acc0[r];
    lds_logits[row * 64 + 16 + ncol] = acc1[r];
    lds_logits[row * 64 + 32 + ncol] = acc2[r];
    lds_logits[row * 64 + 48 + ncol] = acc3[r];
  }
  __syncthreads();

  // Phase 2a: one token per thread (tid < 64): softmax stats, argmax, gate.
  if (tid < 64) {
    const float* row = &lds_logits[tid * 64];
    float mx = row[0];
    int amax = 0;
    #pragma unroll 4
    for (int e = 1; e < E_; ++e) {
      float v = row[e];
      if (v > mx) { mx = v; amax = e; }   // strict '>' => first-index ties
    }
    float sum = 0.0f;
    #pragma unroll 4
    for (int e = 0; e < E_; ++e) sum += expf(row[e] - mx);
    const float rsum = 1.0f / sum;
    const int t = blockTok + tid;
    expert_out[t] = amax;
    gate_out[t]   = rsum;                 // gate = exp(mx-mx)/sum = 1/sum
    lds_max[tid]  = mx;
    lds_rsum[tid] = rsum;
  }
  __syncthreads();

  // Phase 2b: per-expert sums of softmax probs across this block's 64 tokens.
  if (tid < 64) {
    const int e = tid;
    float s = 0.0f;
    #pragma unroll 4
    for (int t = 0; t < 64; ++t)
      s += expf(lds_logits[t * 64 + e] - lds_max[t]) * lds_rsum[t];
    const int g = blockTok / S_;
    atomicAdd(&proxy_sum[g * E_ + e], s);
  }
}

// ---------------------------------------------------------------------------
// Kernel 2: stable position-in-expert scan. One wave32 per group; per 32-token
// chunk, ballot-match lanes with equal expert, rank within chunk, LDS counters.
// Also emits per-(g,e) total counts (pre-capacity) for the aux loss.
// ---------------------------------------------------------------------------
__global__ __launch_bounds__(32) void k_scan(const int* __restrict__ expert_in,
                                             int*   __restrict__ pos_out,
                                             float* __restrict__ counts) {
  __shared__ int cnt[E_];
  const int lane = threadIdx.x;
  const int g = blockIdx.x;
  cnt[lane] = 0;
  cnt[lane + 32] = 0;
  __syncthreads();

  for (int s0 = 0; s0 < S_; s0 += 32) {
    const int t = g * S_ + s0 + lane;
    const int e = expert_in[t];
    unsigned long long eq = ~0ull;
    #pragma unroll
    for (int b = 0; b < 6; ++b) {
      const unsigned long long m = __ballot((e >> b) & 1);
      eq &= ((e >> b) & 1) ? m : ~m;
    }
    eq &= 0xFFFFFFFFull;                                // wave32 lanes only
    const int within = __popcll(eq & ((1ull << lane) - 1ull));
    const int total  = __popcll(eq);
    const int base   = cnt[e];
    const int pos    = base + within;
    __syncthreads();
    if (within == 0) cnt[e] = base + total;             // group leader updates
    __syncthreads();
    pos_out[t] = (pos < CAP_) ? pos : -1;               // capacity drop
  }
  counts[g * E_ + lane]      = (float)cnt[lane];
  counts[g * E_ + lane + 32] = (float)cnt[lane + 32];
}

// ---------------------------------------------------------------------------
// Kernel 3: zero-fill both big outputs (671 MB) with non-temporal 128b stores.
// ---------------------------------------------------------------------------
__global__ __launch_bounds__(256) void k_fill(v4f* __restrict__ out, size_t n4) {
  const v4f z = {0.f, 0.f, 0.f, 0.f};
  for (size_t i = (size_t)blockIdx.x * blockDim.x + threadIdx.x; i < n4;
       i += (size_t)gridDim.x * blockDim.x)
    __builtin_nontemporal_store(z, &out[i]);
}

// ---------------------------------------------------------------------------
// Kernel 4: scatter the <=8192 nonzeros into combine_tensor / dispatch_mask.
// ---------------------------------------------------------------------------
__global__ __launch_bounds__(256) void k_scatter(const int* __restrict__ expert,
                                                 const float* __restrict__ gate,
                                                 const int* __restrict__ pos,
                                                 float* __restrict__ out,
                                                 size_t gsec) {
  const int t = blockIdx.x * blockDim.x + threadIdx.x;
  if (t >= GS_) return;
  const int p = pos[t];
  if (p < 0) return;
  const int e = expert[t];
  const float gv = gate[t];
  const size_t idx = ((size_t)t * E_ + e) * CAP_ + p;
  out[idx] = gv;                                    // combine_tensor
  out[gsec + idx] = (gv != 0.0f) ? 1.0f : 0.0f;     // dispatch_mask
}

// ---------------------------------------------------------------------------
// Kernel 5: aux loss = mean_{g,e}(density_proxy * density) * E*E*coef.
// ---------------------------------------------------------------------------
__global__ __launch_bounds__(256) void k_aux(const float* __restrict__ counts,
                                             const float* __restrict__ proxy,
                                             float* __restrict__ out_aux) {
  __shared__ float red[256];
  const int t = threadIdx.x;                        // t = g*E + e, 256 entries
  const float denom = 1.0f + 1e-6f;
  const float d1 = counts[t] / (float)S_ / denom;
  const float dp = proxy[t]  / (float)S_ / denom;
  red[t] = d1 * dp;
  __syncthreads();
  for (int off = 128; off > 0; off >>= 1) {
    if (t < off) red[t] += red[t + off];
    __syncthreads();
  }
  if (t == 0)
    out_aux[0] = red[0] * ((float)(E_ * E_) * 0.01f / (float)(G_ * E_));
}

__global__ __launch_bounds__(256) void k_init(float* __restrict__ proxy) {
  proxy[threadIdx.x] = 0.0f;                        // G*E == 256
}

// ---------------------------------------------------------------------------
extern "C" void kernel_launch(void* const* d_in, const int* in_sizes, int n_in,
                              void* d_out, int out_size, void* d_ws, size_t ws_size,
                              hipStream_t stream) {
  const float* inp = (const float*)d_in[0];         // (4,2048,1024) f32
  const float* W   = (const float*)d_in[1];         // (1024,64) f32
  float* out = (float*)d_out;                       // combine | dispatch | aux

  // Workspace layout
  int*   d_expert = (int*)d_ws;                     // GS ints
  float* d_gate   = (float*)(d_expert + GS_);       // GS floats
  int*   d_pos    = (int*)(d_gate + GS_);           // GS ints
  float* d_counts = (float*)(d_pos + GS_);          // G*E floats
  float* d_proxy  = d_counts + G_ * E_;             // G*E floats

  k_init<<<1, 256, 0, stream>>>(d_proxy);
  k_gate<<<GS_ / 64, 128, 0, stream>>>(inp, W, d_expert, d_gate, d_proxy);
  k_scan<<<G_, 32, 0, stream>>>(d_expert, d_pos, d_counts);
  k_fill<<<4096, 256, 0, stream>>>((v4f*)out, (2 * GSEC_) / 4);
  k_scatter<<<GS_ / 256, 256, 0, stream>>>(d_expert, d_gate, d_pos, out, GSEC_);
  k_aux<<<1, 256, 0, stream>>>(d_counts, d_proxy, out + 2 * GSEC_);
}